// center_based_mapping_layer_26147760898846
// MI455X (gfx1250) — compile-verified
//
#include <hip/hip_runtime.h>
#include <hip/hip_bf16.h>
#include <math.h>

typedef float v2f __attribute__((ext_vector_type(2)));
typedef float v8f __attribute__((ext_vector_type(8)));

#define IN_CH  64
#define OUT_CH 32
#define S_TOT  (32 * 48 * 48)   // 73728 spatial sites
#define WAVES_PER_BLOCK 8
#define S_PER_BLOCK (WAVES_PER_BLOCK * 16)   // 128
#define NBLOCKS (S_TOT / S_PER_BLOCK)        // 576, exact

// ---------------------------------------------------------------------------
// Prep kernel (tiny): fold dim_scale and the -2 factor into B, precompute r[c].
//   Bws[k*32 + c] = -2 * w[k] * center[c*64 + k]
//   rws[c]        = sum_k w[k] * center[c*64 + k]^2
// ---------------------------------------------------------------------------
__global__ void cbm_prep_kernel(const float* __restrict__ center,
                                const float* __restrict__ dim_scale,
                                float* __restrict__ Bws,
                                float* __restrict__ rws) {
    int t = threadIdx.x;
    if (t < IN_CH) {
        float m2w = -2.0f * dim_scale[t];
        #pragma unroll
        for (int c = 0; c < OUT_CH; ++c)
            Bws[t * OUT_CH + c] = m2w * center[c * IN_CH + t];
    } else if (t < IN_CH + OUT_CH) {
        int c = t - IN_CH;
        float r = 0.0f;
        #pragma unroll
        for (int k = 0; k < IN_CH; ++k) {
            float cc = center[c * IN_CH + k];
            r += dim_scale[k] * cc * cc;
        }
        rws[c] = r;
    }
}

// ---------------------------------------------------------------------------
// Main kernel: per wave, a 16-site spatial tile x all 32 classes via two
// f32 WMMA accumulator chains (K = 64 in 16 steps of K=4).
// ---------------------------------------------------------------------------
__global__ __launch_bounds__(256) void cbm_main_kernel(
    const float* __restrict__ X,          // [64][S] channel-major input
    const float* __restrict__ Bws,        // [64][32]  (-2*w*center)
    const float* __restrict__ rws,        // [32]
    const float* __restrict__ dim_scale,  // [64]
    const float* __restrict__ cls_scale,  // [32]
    float* __restrict__ out)              // [S][32] flat
{
    __shared__ float Bs[IN_CH * OUT_CH];          // 8 KB
    __shared__ float wls[IN_CH];
    __shared__ float rls[OUT_CH];
    __shared__ float sls[OUT_CH];
    __shared__ float qls[WAVES_PER_BLOCK * 16];

    const int tid = threadIdx.x;

    // Stage B / scales into LDS (block-cooperative, coalesced).
    #pragma unroll
    for (int i = 0; i < (IN_CH * OUT_CH) / 256; ++i)
        Bs[tid + i * 256] = Bws[tid + i * 256];
    if (tid < IN_CH)                       wls[tid] = dim_scale[tid];
    else if (tid < IN_CH + OUT_CH)         rls[tid - IN_CH] = rws[tid - IN_CH];
    else if (tid < IN_CH + 2 * OUT_CH)     sls[tid - IN_CH - OUT_CH] = cls_scale[tid - IN_CH - OUT_CH];
    __syncthreads();

    const int lane = tid & 31;
    const int wave = tid >> 5;
    const int half = lane >> 4;       // 0: K pair {k0,k0+1}; 1: {k0+2,k0+3}
    const int m    = lane & 15;
    const int s0   = (blockIdx.x * WAVES_PER_BLOCK + wave) * 16;

    v8f acc0 = {};   // classes 0..15
    v8f acc1 = {};   // classes 16..31
    float qp = 0.0f; // per-lane partial of q[s0+m] over this lane's K values

    #pragma unroll
    for (int step = 0; step < 16; ++step) {
        const int kk = step * 4 + half * 2;
        const float* px = X + (size_t)kk * S_TOT + s0 + m;

        // A fragment: lane holds x[s0+m, kk] and x[s0+m, kk+1] (16x4 f32 layout)
        v2f a;
        a.x = px[0];
        a.y = px[S_TOT];
        if (step < 15)
            __builtin_prefetch(px + 4 * (size_t)S_TOT, 0, 3);   // next k-step lines

        // q partial (dim_scale-weighted squares)
        float w0 = wls[kk], w1 = wls[kk + 1];
        qp = fmaf(w0 * a.x, a.x, qp);
        qp = fmaf(w1 * a.y, a.y, qp);

        // B fragments from LDS: row kk / kk+1, classes m and 16+m
        v2f b0, b1;
        b0.x = Bs[kk * OUT_CH + m];
        b0.y = Bs[(kk + 1) * OUT_CH + m];
        b1.x = Bs[kk * OUT_CH + 16 + m];
        b1.y = Bs[(kk + 1) * OUT_CH + 16 + m];

        acc0 = __builtin_amdgcn_wmma_f32_16x16x4_f32(false, a, false, b0,
                                                     (short)0, acc0, false, false);
        acc1 = __builtin_amdgcn_wmma_f32_16x16x4_f32(false, a, false, b1,
                                                     (short)0, acc1, false, false);
    }

    // Combine the two half-wave K partials of q[s0+m], publish per-row via LDS
    float qfull = qp + __shfl_xor(qp, 16, 32);
    if (lane < 16) qls[wave * 16 + m] = qfull;
    __syncthreads();

    const float rA = rls[m],      rB = rls[16 + m];
    const float sA = sls[m],      sB = sls[16 + m];

    // C/D fragment layout: VGPR j <-> row M = j + 8*half, N = lane&15
    #pragma unroll
    for (int j = 0; j < 8; ++j) {
        const int   sm  = j + half * 8;
        const float q   = qls[wave * 16 + sm];
        const size_t row = (size_t)(s0 + sm) * OUT_CH;

        float dA = (q + acc0[j] + rA) * (1.0f / 64.0f);
        float dB = (q + acc1[j] + rB) * (1.0f / 64.0f);
        out[row + m]      = sA * sqrtf(1e-6f + dA);
        out[row + 16 + m] = sB * sqrtf(1e-6f + dB);
    }
}

// ---------------------------------------------------------------------------
extern "C" void kernel_launch(void* const* d_in, const int* in_sizes, int n_in,
                              void* d_out, int out_size, void* d_ws, size_t ws_size,
                              hipStream_t stream) {
    (void)in_sizes; (void)n_in; (void)out_size; (void)ws_size;

    const float* in_tensor = (const float*)d_in[0];  // [1,64,32,48,48]
    const float* center    = (const float*)d_in[1];  // [1,1,32,64]
    const float* dim_scale = (const float*)d_in[2];  // [1,1,1,64]
    const float* cls_scale = (const float*)d_in[3];  // [1,1,32]
    float*       out       = (float*)d_out;          // [1,32,32,48,48] flat = [S,32]

    float* Bws = (float*)d_ws;                 // 64*32 floats = 8 KB
    float* rws = Bws + IN_CH * OUT_CH;         // 32 floats

    cbm_prep_kernel<<<1, IN_CH + OUT_CH, 0, stream>>>(center, dim_scale, Bws, rws);
    cbm_main_kernel<<<NBLOCKS, 256, 0, stream>>>(in_tensor, Bws, rws,
                                                 dim_scale, cls_scale, out);
}